// RetraceLoss_16200616641135
// MI455X (gfx1250) — compile-verified
//
#include <hip/hip_runtime.h>
#include <hip/hip_bf16.h>

// Retrace loss, T=2048, B=8192, fp32 — bandwidth-bound streaming scan.
// pass1: reads all 5 inputs exactly once via double-buffered
//        global_load_async_to_lds_b128 (CDNA5 ASYNCcnt path), reducing each
//        (chunk, 4-column) to an affine transform + quadratic loss coeffs.
// pass2: chains the 32 chunk transforms per column -> per-column SSE.
// pass3: sums 8192 column SSEs with V_WMMA_F32_16X16X4_F32 (ones-matrix
//        reduction), fixed-order finish -> deterministic scalar mean.

#define GAMMA_F 0.95f
constexpr int T  = 2048;
constexpr int B  = 8192;
constexpr int K  = 32;        // chunks along T
constexpr int CH = T / K;     // 64 timesteps per chunk
constexpr int BV = B / 4;     // float4 column groups

typedef float v2f __attribute__((ext_vector_type(2)));
typedef float v8f __attribute__((ext_vector_type(8)));

__device__ __forceinline__ void async_copy16(unsigned lds_off, const float* gp) {
  // gfx1250 async global -> LDS, 16 bytes per lane, tracked by ASYNCcnt
  asm volatile("global_load_async_to_lds_b128 %0, %1, off"
               :: "v"(lds_off), "v"(gp) : "memory");
}

__device__ __forceinline__ void retrace_step(
    float sa, float tg, float r, float lj, float lk,
    float& u, float& v, float& P,
    float& Sdd, float& Sde, float& Sdf,
    float& See, float& Sef, float& Sff)
{
  float c = GAMMA_F * expf(fminf(lk - lj, 0.0f));
  u = fmaf(c, u, 1.0f);            // U[t] = 1 + c*U[t+1]
  v = fmaf(c, v, r - tg);          // V[t] = (r - target) + c*V[t+1]
  P = c * P;                       // running product of c over chunk
  float d = sa - fmaf(tg, u, v);   // zero-carry residual
  float e = tg * P;                // dU_in coefficient
  float f = P;                     // dV_in coefficient
  Sdd = fmaf(d, d, Sdd);
  Sde = fmaf(d, e, Sde);
  Sdf = fmaf(d, f, Sdf);
  See = fmaf(e, e, See);
  Sef = fmaf(e, f, Sef);
  Sff = fmaf(f, f, Sff);
}

__global__ __launch_bounds__(256) void retrace_pass1(
    const float* __restrict__ sav, const float* __restrict__ tgt,
    const float* __restrict__ rew, const float* __restrict__ ltraj,
    const float* __restrict__ ltask, float* __restrict__ ws)
{
  // [stage][array][tid*4 + j] : 2*5*1024 floats = 40 KB
  __shared__ __align__(16) float smem[2][5][1024];

  const int tid  = threadIdx.x;
  const int gid  = blockIdx.x * blockDim.x + tid;
  const int col4 = gid & (BV - 1);   // group of 4 consecutive columns
  const int k    = gid / BV;         // chunk index along T
  const int b    = col4 * 4;
  const int tlo  = k * CH;
  const int thi  = tlo + CH - 1;

  // byte offset of this thread's slice in LDS; per (stage,array) stride 4096 B
  const unsigned tb = (unsigned)(size_t)(const void*)&smem[0][0][tid * 4];

  float4 z = make_float4(0.f, 0.f, 0.f, 0.f);
  float4 u = z, v = z;
  float4 P = make_float4(1.f, 1.f, 1.f, 1.f);
  float4 Sdd = z, Sde = z, Sdf = z, See = z, Sef = z, Sff = z;

  // preload stage for t = thi
  {
    const size_t off0 = (size_t)thi * B + b;
    const unsigned sb = tb + (unsigned)((thi & 1) * 5) * 4096u;
    async_copy16(sb + 0u * 4096u, sav   + off0);
    async_copy16(sb + 1u * 4096u, tgt   + off0);
    async_copy16(sb + 2u * 4096u, rew   + off0);
    async_copy16(sb + 3u * 4096u, ltraj + off0);
    async_copy16(sb + 4u * 4096u, ltask + off0);
  }

  for (int t = thi; t >= tlo; --t) {
    const int cur = t & 1;
    const size_t off = (size_t)t * B + b;

    if (t >= 8) {
      // speculative L2 prefetch well ahead of the async pipeline
      const size_t poff = off - (size_t)8 * B;
      __builtin_prefetch(sav   + poff, 0, 1);
      __builtin_prefetch(tgt   + poff, 0, 1);
      __builtin_prefetch(rew   + poff, 0, 1);
      __builtin_prefetch(ltraj + poff, 0, 1);
      __builtin_prefetch(ltask + poff, 0, 1);
    }

    if (t > tlo) {
      // issue next stage, then wait only for the current stage (<=5 pending)
      const size_t noff = off - (size_t)B;
      const unsigned nb = tb + (unsigned)(((t - 1) & 1) * 5) * 4096u;
      async_copy16(nb + 0u * 4096u, sav   + noff);
      async_copy16(nb + 1u * 4096u, tgt   + noff);
      async_copy16(nb + 2u * 4096u, rew   + noff);
      async_copy16(nb + 3u * 4096u, ltraj + noff);
      async_copy16(nb + 4u * 4096u, ltask + noff);
      asm volatile("s_wait_asynccnt 0x5" ::: "memory");
    } else {
      asm volatile("s_wait_asynccnt 0x0" ::: "memory");
    }

    const float4 sa = *(const float4*)&smem[cur][0][tid * 4];
    const float4 tg = *(const float4*)&smem[cur][1][tid * 4];
    const float4 r  = *(const float4*)&smem[cur][2][tid * 4];
    const float4 lj = *(const float4*)&smem[cur][3][tid * 4];
    const float4 lk = *(const float4*)&smem[cur][4][tid * 4];

    retrace_step(sa.x, tg.x, r.x, lj.x, lk.x, u.x, v.x, P.x, Sdd.x, Sde.x, Sdf.x, See.x, Sef.x, Sff.x);
    retrace_step(sa.y, tg.y, r.y, lj.y, lk.y, u.y, v.y, P.y, Sdd.y, Sde.y, Sdf.y, See.y, Sef.y, Sff.y);
    retrace_step(sa.z, tg.z, r.z, lj.z, lk.z, u.z, v.z, P.z, Sdd.z, Sde.z, Sdf.z, See.z, Sef.z, Sff.z);
    retrace_step(sa.w, tg.w, r.w, lj.w, lk.w, u.w, v.w, P.w, Sdd.w, Sde.w, Sdf.w, See.w, Sef.w, Sff.w);
  }

  // ws layout: 9 planes of [K*B] floats: A, bu, bv, Sdd, Sde, Sdf, See, Sef, Sff
  const size_t N  = (size_t)K * B;
  const size_t i4 = ((size_t)k * B + b) / 4;
  ((float4*)(ws + 0 * N))[i4] = P;     // A  = prod(c) over chunk
  ((float4*)(ws + 1 * N))[i4] = u;     // bu = u at chunk start
  ((float4*)(ws + 2 * N))[i4] = v;     // bv = v at chunk start
  ((float4*)(ws + 3 * N))[i4] = Sdd;
  ((float4*)(ws + 4 * N))[i4] = Sde;
  ((float4*)(ws + 5 * N))[i4] = Sdf;
  ((float4*)(ws + 6 * N))[i4] = See;
  ((float4*)(ws + 7 * N))[i4] = Sef;
  ((float4*)(ws + 8 * N))[i4] = Sff;
}

__global__ __launch_bounds__(256) void retrace_pass2(
    const float* __restrict__ ws, float* __restrict__ colsse)
{
  const int b = blockIdx.x * 256 + threadIdx.x;   // one thread per column
  const size_t N = (size_t)K * B;
  float U = 0.f, V = 0.f, sse = 0.f;
  for (int k = K - 1; k >= 0; --k) {              // chain chunks from t=T downward
    const size_t i = (size_t)k * B + b;
    const float A   = ws[0 * N + i];
    const float bu  = ws[1 * N + i];
    const float bv  = ws[2 * N + i];
    const float Sdd = ws[3 * N + i];
    const float Sde = ws[4 * N + i];
    const float Sdf = ws[5 * N + i];
    const float See = ws[6 * N + i];
    const float Sef = ws[7 * N + i];
    const float Sff = ws[8 * N + i];
    // sum over chunk of (d - e*U - f*V)^2 via precomputed quadratic form
    sse += Sdd - 2.f * (U * Sde + V * Sdf)
         + U * (U * See + 2.f * V * Sef) + V * V * Sff;
    const float Un = fmaf(A, U, bu);
    V = fmaf(A, V, bv);
    U = Un;
  }
  colsse[b] = sse;
}

__global__ void retrace_pass3(const float* __restrict__ colsse,
                              float* __restrict__ out)
{
#if __has_builtin(__builtin_amdgcn_wmma_f32_16x16x4_f32)
  // Single wave32. D = A(16x4) * ones(4x16) + C accumulates total row-sums;
  // every loaded value lands in exactly one A slot, so C row-sums = grand sum.
  const int lane = (int)threadIdx.x;      // 0..31, all lanes active (EXEC all 1s)
  v8f acc = {0.f, 0.f, 0.f, 0.f, 0.f, 0.f, 0.f, 0.f};
  const v2f ones = {1.f, 1.f};
  for (int base = 0; base < B; base += 64) {
    v2f a;
    a.x = colsse[base + lane];
    a.y = colsse[base + 32 + lane];
    acc = __builtin_amdgcn_wmma_f32_16x16x4_f32(
        /*neg_a=*/false, a, /*neg_b=*/false, ones,
        /*c_mod=*/(short)0, acc, /*reuse_a=*/false, /*reuse_b=*/false);
  }
  // D[m][n] identical over n; column n=0 lives in lanes 0 (m=r) and 16 (m=r+8)
  __shared__ float cbuf[8][32];
  #pragma unroll
  for (int r = 0; r < 8; ++r) cbuf[r][lane] = acc[r];
  __syncthreads();
  if (lane == 0) {
    float s = 0.f;
    #pragma unroll
    for (int r = 0; r < 8; ++r) s += cbuf[r][0] + cbuf[r][16];
    out[0] = s / (float)((size_t)T * (size_t)B);
  }
#else
  if (threadIdx.x == 0) {
    float s = 0.f;
    for (int i = 0; i < B; ++i) s += colsse[i];
    out[0] = s / (float)((size_t)T * (size_t)B);
  }
#endif
}

extern "C" void kernel_launch(void* const* d_in, const int* in_sizes, int n_in,
                              void* d_out, int out_size, void* d_ws, size_t ws_size,
                              hipStream_t stream) {
  const float* sav   = (const float*)d_in[0];  // state_action_values
  const float* tgt   = (const float*)d_in[1];  // target_state_action_values
  const float* rew   = (const float*)d_in[2];  // rewards
  const float* ltraj = (const float*)d_in[3];  // log_trajectory_action_probs
  const float* ltask = (const float*)d_in[4];  // log_task_action_probs

  float* ws     = (float*)d_ws;
  float* colsse = ws + (size_t)9 * K * B;      // B per-column SSEs after 9 planes
  float* out    = (float*)d_out;

  retrace_pass1<<<(BV * K) / 256, 256, 0, stream>>>(sav, tgt, rew, ltraj, ltask, ws);
  retrace_pass2<<<B / 256, 256, 0, stream>>>(ws, colsse);
  retrace_pass3<<<1, 32, 0, stream>>>(colsse, out);
}